// VariationalQuantumCircuit_3882650436750
// MI455X (gfx1250) — compile-verified
//
#include <hip/hip_runtime.h>
#include <math.h>

// ---------------------------------------------------------------------------
// 24-qubit variational circuit simulator for gfx1250 (MI455X).
//   - state = 2^24 complex64 (float2 interleaved), 134 MB: fits in 192MB L2.
//   - per layer: 6 fused 4-qubit gates, each a 16x16 complex unitary applied
//     with v_wmma_f32_16x16x4_f32 (4 real 16x16x16 matmuls = 16 WMMA / tile).
//   - CNOT chain between layers collapses to the index permutation
//     y -> y ^ ((y<<1) & 0xFFFFFF), fused into the next layer's first pass.
// ---------------------------------------------------------------------------

#define N_QUBITS 24
#define N_STATES (1u << N_QUBITS)
#define MASK24   0x00FFFFFFu

typedef float v2f __attribute__((ext_vector_type(2)));
typedef float v8f __attribute__((ext_vector_type(8)));

#if defined(__has_builtin)
#if __has_builtin(__builtin_amdgcn_wmma_f32_16x16x4_f32)
#define HAVE_WMMA_F32 1
#else
#define HAVE_WMMA_F32 0
#endif
#else
#define HAVE_WMMA_F32 0
#endif

enum { MODE_INPLACE = 0, MODE_PERM = 1, MODE_INIT = 2 };

__device__ __forceinline__ float2 cmul(float2 a, float2 b) {
  return make_float2(a.x * b.x - a.y * b.y, a.x * b.y + a.y * b.x);
}
__device__ __forceinline__ float2 cadd(float2 a, float2 b) {
  return make_float2(a.x + b.x, a.y + b.y);
}

// --------------------------- norm reduction --------------------------------

__global__ __launch_bounds__(256) void vqc_norm_partial(
    const float* __restrict__ re, const float* __restrict__ im,
    float* __restrict__ partial) {
  __shared__ float sm[256];
  const int t = threadIdx.x;
  size_t i = (size_t)blockIdx.x * 2048 + t;  // 8192 blocks * 2048 = 2^24
  float s = 0.f;
#pragma unroll
  for (int k = 0; k < 8; ++k) {
    float a = re[i], b = im[i];
    s += a * a + b * b;
    i += 256;
  }
  sm[t] = s;
  __syncthreads();
  for (int w = 128; w > 0; w >>= 1) {
    if (t < w) sm[t] += sm[t + w];
    __syncthreads();
  }
  if (t == 0) partial[blockIdx.x] = sm[0];
}

__global__ __launch_bounds__(256) void vqc_norm_final(
    const float* __restrict__ partial, float* __restrict__ invn) {
  __shared__ double sm[256];
  const int t = threadIdx.x;
  double s = 0.0;
  for (int k = t; k < 8192; k += 256) s += (double)partial[k];
  sm[t] = s;
  __syncthreads();
  for (int w = 128; w > 0; w >>= 1) {
    if (t < w) sm[t] += sm[t + w];
    __syncthreads();
  }
  if (t == 0) *invn = (float)(1.0 / sqrt(sm[0]));
}

// --------------------------- gate construction -----------------------------

// One fused 2x2 unitary per (layer, qubit): U = Rz(p2) * Ry(p1) * Rx(p0).
__global__ void vqc_build_u2(const float* __restrict__ params,
                             float2* __restrict__ u2) {
  const int t = threadIdx.x;
  if (t >= 96) return;
  const float a0 = params[3 * t + 0] * 0.5f;
  const float a1 = params[3 * t + 1] * 0.5f;
  const float a2 = params[3 * t + 2] * 0.5f;
  const float cx = cosf(a0), sx = sinf(a0);
  const float cy = cosf(a1), sy = sinf(a1);
  const float cz = cosf(a2), sz = sinf(a2);
  // M = Ry * Rx
  const float2 M00 = make_float2(cy * cx,  sy * sx);
  const float2 M01 = make_float2(-sy * cx, -cy * sx);
  const float2 M10 = make_float2(sy * cx,  -cy * sx);
  const float2 M11 = make_float2(cy * cx,  -sy * sx);
  const float2 z0 = make_float2(cz, -sz);  // Rz diag entries
  const float2 z1 = make_float2(cz,  sz);
  u2[4 * t + 0] = cmul(z0, M00);  // [row*2 + col]
  u2[4 * t + 1] = cmul(z0, M01);
  u2[4 * t + 2] = cmul(z1, M10);
  u2[4 * t + 3] = cmul(z1, M11);
}

// 16x16 unitary per (layer, group-of-4 qubits): Kronecker of four 2x2s.
// Row/col bit j corresponds to qubit 4*grp + j (little-endian).
__global__ __launch_bounds__(256) void vqc_build_u4(
    const float2* __restrict__ u2, float2* __restrict__ g4) {
  const int g = blockIdx.x;   // 0..23 : layer*6 + grp
  const int t = threadIdx.x;  // 0..255 : r*16 + c
  const int r = t >> 4, c = t & 15;
  const int layer = g / 6, grp = g % 6;
  float2 prod = make_float2(1.f, 0.f);
#pragma unroll
  for (int j = 0; j < 4; ++j) {
    const int q = layer * 24 + grp * 4 + j;
    const int rb = (r >> j) & 1, cb = (c >> j) & 1;
    prod = cmul(prod, u2[4 * q + 2 * rb + cb]);
  }
  g4[g * 256 + t] = prod;
}

// --------------------------- 4-qubit gate pass -----------------------------
// State reshaped as [high][16 group bits][2^p low]. Column index
// col = high*2^p + low (2^20 columns). Each wave handles one 16-column tile:
// D(16x16) = U(16x16 complex) * X(16x16 complex), via fp32 WMMA.

#if HAVE_WMMA_F32
__global__ __launch_bounds__(256) void vqc_gate4_wmma(
    const float2* in, float2* out, const float2* __restrict__ U,
    const float* __restrict__ re, const float* __restrict__ im,
    const float* __restrict__ invn, int p, int mode) {
  const int tid  = threadIdx.x;
  const int lane = tid & 31;
  const int wave = tid >> 5;
  const unsigned unit = blockIdx.x * 8u + (unsigned)wave;  // 65536 units
  const int r = lane & 15;          // A row / B-D column within tile
  const int e = lane >> 4;          // half-select per ISA VGPR layouts
  const unsigned col  = unit * 16u + (unsigned)r;
  const unsigned low  = col & ((1u << p) - 1u);
  const unsigned high = col >> p;
  const unsigned base = (high << (p + 4)) | low;  // element index for b=0

  // A operands: U row r, columns {4j+2e, 4j+2e+1}. Pre-negate Ui for Dr.
  v2f Ar[4], Ai[4], Ain[4];
#pragma unroll
  for (int j = 0; j < 4; ++j) {
    const int c0 = 4 * j + 2 * e;
    const float2 g0 = U[r * 16 + c0];
    const float2 g1 = U[r * 16 + c0 + 1];
    Ar[j][0] = g0.x;  Ar[j][1] = g1.x;
    Ai[j][0] = g0.y;  Ai[j][1] = g1.y;
    Ain[j][0] = -g0.y; Ain[j][1] = -g1.y;
  }

  const float inv = (mode == MODE_INIT) ? *invn : 1.0f;

  // B operands: column `col`, rows {4j+2e, 4j+2e+1}.
  v2f Br[4], Bi[4];
#pragma unroll
  for (int j = 0; j < 4; ++j) {
    const unsigned b0 = 4u * j + 2u * e;
    const unsigned i0 = base | (b0 << p);
    const unsigned i1 = base | ((b0 + 1u) << p);
    float2 x0, x1;
    if (mode == MODE_INIT) {
      x0 = make_float2(re[i0] * inv, im[i0] * inv);
      x1 = make_float2(re[i1] * inv, im[i1] * inv);
    } else if (mode == MODE_PERM) {
      const unsigned s0 = (i0 ^ (i0 << 1)) & MASK24;  // fused CNOT chain
      const unsigned s1 = (i1 ^ (i1 << 1)) & MASK24;
      x0 = in[s0];
      x1 = in[s1];
    } else {
      x0 = in[i0];
      x1 = in[i1];
    }
    Br[j][0] = x0.x;  Br[j][1] = x1.x;
    Bi[j][0] = x0.y;  Bi[j][1] = x1.y;
  }

  v8f accR = {0.f, 0.f, 0.f, 0.f, 0.f, 0.f, 0.f, 0.f};
  v8f accI = {0.f, 0.f, 0.f, 0.f, 0.f, 0.f, 0.f, 0.f};
  // Dr = Ur*Xr + (-Ui)*Xi
#pragma unroll
  for (int j = 0; j < 4; ++j)
    accR = __builtin_amdgcn_wmma_f32_16x16x4_f32(false, Ar[j], false, Br[j],
                                                 (short)0, accR, false, false);
#pragma unroll
  for (int j = 0; j < 4; ++j)
    accR = __builtin_amdgcn_wmma_f32_16x16x4_f32(false, Ain[j], false, Bi[j],
                                                 (short)0, accR, false, false);
  // Di = Ur*Xi + Ui*Xr
#pragma unroll
  for (int j = 0; j < 4; ++j)
    accI = __builtin_amdgcn_wmma_f32_16x16x4_f32(false, Ar[j], false, Bi[j],
                                                 (short)0, accI, false, false);
#pragma unroll
  for (int j = 0; j < 4; ++j)
    accI = __builtin_amdgcn_wmma_f32_16x16x4_f32(false, Ai[j], false, Br[j],
                                                 (short)0, accI, false, false);

  // D layout: lane holds rows {8e .. 8e+7} of its column.
#pragma unroll
  for (int j = 0; j < 8; ++j) {
    const unsigned row = 8u * e + (unsigned)j;
    out[base | (row << p)] = make_float2(accR[j], accI[j]);
  }
}
#else
// Scalar fallback: one thread owns one column (16 amplitudes).
__global__ __launch_bounds__(256) void vqc_gate4_scalar(
    const float2* in, float2* out, const float2* __restrict__ U,
    const float* __restrict__ re, const float* __restrict__ im,
    const float* __restrict__ invn, int p, int mode) {
  const unsigned col  = blockIdx.x * 256u + threadIdx.x;  // 2^20 columns
  const unsigned low  = col & ((1u << p) - 1u);
  const unsigned high = col >> p;
  const unsigned base = (high << (p + 4)) | low;
  const float inv = (mode == MODE_INIT) ? *invn : 1.0f;
  float2 x[16];
#pragma unroll
  for (int b = 0; b < 16; ++b) {
    const unsigned i = base | ((unsigned)b << p);
    if (mode == MODE_INIT) {
      x[b] = make_float2(re[i] * inv, im[i] * inv);
    } else if (mode == MODE_PERM) {
      const unsigned s = (i ^ (i << 1)) & MASK24;
      x[b] = in[s];
    } else {
      x[b] = in[i];
    }
  }
#pragma unroll
  for (int a = 0; a < 16; ++a) {
    float2 acc = make_float2(0.f, 0.f);
#pragma unroll
    for (int b = 0; b < 16; ++b) acc = cadd(acc, cmul(U[a * 16 + b], x[b]));
    out[base | ((unsigned)a << p)] = acc;
  }
}
#endif

static inline void launch_gate(const float2* in, float2* out, const float2* U,
                               const float* re, const float* im,
                               const float* invn, int p, int mode,
                               hipStream_t s) {
#if HAVE_WMMA_F32
  vqc_gate4_wmma<<<8192, 256, 0, s>>>(in, out, U, re, im, invn, p, mode);
#else
  vqc_gate4_scalar<<<4096, 256, 0, s>>>(in, out, U, re, im, invn, p, mode);
#endif
}

// ---------------------------------------------------------------------------

extern "C" void kernel_launch(void* const* d_in, const int* in_sizes, int n_in,
                              void* d_out, int out_size, void* d_ws,
                              size_t ws_size, hipStream_t stream) {
  (void)in_sizes; (void)n_in; (void)out_size; (void)ws_size;
  const float* params = (const float*)d_in[0];
  const float* re     = (const float*)d_in[1];
  const float* im     = (const float*)d_in[2];
  float2* O = (float2*)d_out;  // final state lives here (interleaved re,im)

  // Workspace carve-up (needs ~128 MB + 90 KB).
  char* ws = (char*)d_ws;
  float2* W = (float2*)ws;  // ping-pong state buffer
  size_t off = (size_t)N_STATES * sizeof(float2);
  float* partial = (float*)(ws + off); off += 8192 * sizeof(float);
  float* invn    = (float*)(ws + off); off += 256;
  float2* u2     = (float2*)(ws + off); off += 96 * 4 * sizeof(float2);
  float2* g4     = (float2*)(ws + off);  // 24 * 256 * 8 B

  // Normalization factor + gate matrices.
  vqc_norm_partial<<<8192, 256, 0, stream>>>(re, im, partial);
  vqc_norm_final<<<1, 256, 0, stream>>>(partial, invn);
  vqc_build_u2<<<1, 128, 0, stream>>>(params, u2);
  vqc_build_u4<<<24, 256, 0, stream>>>(u2, g4);

  // Layer 0: first pass reads planar inputs (scaled), then in-place on W.
  launch_gate(nullptr, W, g4 + (0 * 6 + 0) * 256, re, im, invn, 0, MODE_INIT,
              stream);
  for (int g = 1; g < 6; ++g)
    launch_gate(W, W, g4 + (0 * 6 + g) * 256, re, im, invn, 4 * g,
                MODE_INPLACE, stream);

  // Layer 1: CNOT-chain permutation fused into first pass, W -> O.
  launch_gate(W, O, g4 + (1 * 6 + 0) * 256, re, im, invn, 0, MODE_PERM,
              stream);
  for (int g = 1; g < 6; ++g)
    launch_gate(O, O, g4 + (1 * 6 + g) * 256, re, im, invn, 4 * g,
                MODE_INPLACE, stream);

  // Layer 2: perm fused, O -> W.
  launch_gate(O, W, g4 + (2 * 6 + 0) * 256, re, im, invn, 0, MODE_PERM,
              stream);
  for (int g = 1; g < 6; ++g)
    launch_gate(W, W, g4 + (2 * 6 + g) * 256, re, im, invn, 4 * g,
                MODE_INPLACE, stream);

  // Layer 3: perm fused, W -> O. Final state ends in d_out.
  launch_gate(W, O, g4 + (3 * 6 + 0) * 256, re, im, invn, 0, MODE_PERM,
              stream);
  for (int g = 1; g < 6; ++g)
    launch_gate(O, O, g4 + (3 * 6 + g) * 256, re, im, invn, 4 * g,
                MODE_INPLACE, stream);
}